// GaussianEmbeddingDP_16990890623150
// MI455X (gfx1250) — compile-verified
//
#include <hip/hip_runtime.h>

typedef __attribute__((ext_vector_type(2))) float v2f;
typedef __attribute__((ext_vector_type(4))) float v4f;
typedef __attribute__((ext_vector_type(8))) float v8f;

#define TILE_D        128            // row length (floats)
#define ROWS_PER_WAVE 16
#define WAVES_PER_BLK 4
#define ROWS_PER_BLK  (ROWS_PER_WAVE * WAVES_PER_BLK)   // 64
#define ROW_STRIDE    132            // padded LDS row stride (floats): 132*4B mod 256B = 16B
                                     // -> rows land 4 banks apart, conflict-free frag loads

__global__ __launch_bounds__(WAVES_PER_BLK * 32)
void dp_clip_noise_kernel(const float* __restrict__ x,
                          const float* __restrict__ noise,
                          float* __restrict__ out,
                          int n_rows) {
  __shared__ float lds_x[WAVES_PER_BLK * ROWS_PER_WAVE * ROW_STRIDE];
  __shared__ float lds_scale[ROWS_PER_BLK];

  const int lane = threadIdx.x & 31;
  const int wave = threadIdx.x >> 5;
  const long rowbase = (long)blockIdx.x * ROWS_PER_BLK + (long)wave * ROWS_PER_WAVE;
  if (rowbase + ROWS_PER_WAVE > n_rows) return;   // wave-uniform guard (EXEC stays all-1s)

  float* wtile = lds_x + wave * (ROWS_PER_WAVE * ROW_STRIDE);
  const float* gx = x     + rowbase * TILE_D;
  const float* gn = noise + rowbase * TILE_D;
  float*       gy = out   + rowbase * TILE_D;

  // ---- Phase 0: stream 16x128 f32 tile into padded LDS (coalesced b128, NT) ----
#pragma unroll
  for (int t = 0; t < ROWS_PER_WAVE; ++t) {
    v4f v = __builtin_nontemporal_load((const v4f*)(gx + t * TILE_D + 4 * lane));
    *(v4f*)(wtile + t * ROW_STRIDE + 4 * lane) = v;
  }

  // ---- Phase 1: row norms^2 via chained V_WMMA_F32_16X16X4_F32 with B = ones ----
  // A fragment (16x4 f32): lane L<16 holds A[L][k], A[L][k+1]; lane L+16 holds A[L][k+2..k+3]
  v8f acc = {};                       // C = 0
  const v2f bones = {1.0f, 1.0f};    // all-ones B is layout-invariant
  const int arow = lane & 15;
  const int acol = 2 * (lane >> 4);
#pragma unroll
  for (int kk = 0; kk < TILE_D / 4; ++kk) {
    v2f a2 = *(const v2f*)(wtile + arow * ROW_STRIDE + 4 * kk + acol);
    v2f asq = a2 * a2;               // square in VALU, sum along K on the matrix pipe
    acc = __builtin_amdgcn_wmma_f32_16x16x4_f32(
        /*neg_a=*/false, asq, /*neg_b=*/false, bones,
        /*c_mod=*/(short)0, acc, /*reuse_a=*/false, /*reuse_b=*/false);
  }

  // D layout: VGPR r -> row r (lanes 0-15) / row r+8 (lanes 16-31), replicated over N.
  // Lanes 0 and 16 jointly publish all 16 scales to LDS (intra-wave, DS is in-order).
  if ((lane & 15) == 0) {
    const int half = (lane >> 4) * 8;
#pragma unroll
    for (int r = 0; r < 8; ++r) {
      const float n2 = acc[r];
      // scale = 1 / max(norm/clip, 1), clip = 1  ->  min(1, rsqrt(norm^2));  n2==0 -> 1
      lds_scale[wave * ROWS_PER_WAVE + half + r] = fminf(1.0f, __frsqrt_rn(n2));
    }
  }

  // ---- Phase 2: out = x * scale + noise  (b128 streams, NT; x re-read from LDS only) ----
#pragma unroll
  for (int t = 0; t < ROWS_PER_WAVE; ++t) {
    const float s = lds_scale[wave * ROWS_PER_WAVE + t];   // LDS broadcast read
    v4f xv = *(const v4f*)(wtile + t * ROW_STRIDE + 4 * lane);
    v4f nv = __builtin_nontemporal_load((const v4f*)(gn + t * TILE_D + 4 * lane));
    v4f o;
    o.x = fmaf(xv.x, s, nv.x);
    o.y = fmaf(xv.y, s, nv.y);
    o.z = fmaf(xv.z, s, nv.z);
    o.w = fmaf(xv.w, s, nv.w);
    __builtin_nontemporal_store(o, (v4f*)(gy + t * TILE_D + 4 * lane));
  }
}

extern "C" void kernel_launch(void* const* d_in, const int* in_sizes, int n_in,
                              void* d_out, int out_size, void* d_ws, size_t ws_size,
                              hipStream_t stream) {
  const float* x     = (const float*)d_in[0];
  const float* noise = (const float*)d_in[1];
  float*       out   = (float*)d_out;

  const int n_rows = in_sizes[0] / TILE_D;                 // 524288
  const int grid   = (n_rows + ROWS_PER_BLK - 1) / ROWS_PER_BLK;  // 8192
  dp_clip_noise_kernel<<<grid, WAVES_PER_BLK * 32, 0, stream>>>(x, noise, out, n_rows);
}